// RegionLoss_14439680049762
// MI455X (gfx1250) — compile-verified
//
#include <hip/hip_runtime.h>
#include <cstdint>
#include <cstddef>

#define NCLS   80
#define AN     5
#define HH     52
#define WW     52
#define BB     32
#define NBOX   50
#define PLANE  (HH * WW)        // 2704
#define CHN    (NCLS + 5)       // 85
#define RECF   16               // floats per GT record

static __device__ const float d_anchors[AN][2] = {
    {1.3221f, 1.73145f}, {3.19275f, 4.00944f}, {5.05587f, 8.09892f},
    {9.47112f, 4.84053f}, {11.2364f, 10.0071f}};

__device__ inline float sigm(float x) { return 1.0f / (1.0f + expf(-x)); }

__device__ inline float iou_center(float ax, float ay, float aw, float ah,
                                   float bx, float by, float bw, float bh) {
    float ax1 = ax - aw * 0.5f, ax2 = ax + aw * 0.5f;
    float ay1 = ay - ah * 0.5f, ay2 = ay + ah * 0.5f;
    float bx1 = bx - bw * 0.5f, bx2 = bx + bw * 0.5f;
    float by1 = by - bh * 0.5f, by2 = by + bh * 0.5f;
    float iw = fmaxf(fminf(ax2, bx2) - fmaxf(ax1, bx1), 0.0f);
    float ih = fmaxf(fminf(ay2, by2) - fmaxf(ay1, by1), 0.0f);
    float inter = iw * ih;
    float uni = aw * ah + bw * bh - inter;
    return inter / uni;
}

// ---- wave32 sum via the matrix pipe: one V_WMMA_F32_16X16X4_F32 with B=ones
// A layout (16x4 f32): lane m<16 holds A[m][0] (v0) and A[m][1] (v1);
// lane 16+m holds A[m][2], A[m][3]. With v1=0, D[m][*] = x[m] + x[m+16].
// D layout: vgpr r, lane n<16 -> D[r][n]; lane 16+n -> D[r+8][n].
typedef float v2f __attribute__((ext_vector_type(2)));
typedef float v8f __attribute__((ext_vector_type(8)));

__device__ inline float wave_sum(float x) {
#if __has_builtin(__builtin_amdgcn_wmma_f32_16x16x4_f32)
    v2f A; A[0] = x;    A[1] = 0.0f;
    v2f Bv; Bv[0] = 1.0f; Bv[1] = 1.0f;
    v8f C = {0.f, 0.f, 0.f, 0.f, 0.f, 0.f, 0.f, 0.f};
    v8f D = __builtin_amdgcn_wmma_f32_16x16x4_f32(
        false, A, false, Bv, (short)0, C, false, false);
    float s = ((D[0] + D[1]) + (D[2] + D[3])) + ((D[4] + D[5]) + (D[6] + D[7]));
    s += __shfl_xor(s, 16, 32);   // combine rows 0-7 half with rows 8-15 half
    return s;
#else
    for (int off = 16; off > 0; off >>= 1) x += __shfl_xor(x, off, 32);
    return x;
#endif
}

// ---- kernel 0: zero the loss accumulators (graph-safe re-init each launch)
__global__ void k_zero(float* __restrict__ acc) {
    if (threadIdx.x < 8) acc[threadIdx.x] = 0.0f;
}

// ---- kernel 1: per-GT precompute (best anchor, cell, targets, iou_gt)
__global__ void k_pre(const float* __restrict__ pred,
                      const float* __restrict__ target,
                      float* __restrict__ gt) {
    int b = blockIdx.x;
    int n = threadIdx.x;
    if (n >= NBOX) return;
    const float* t = target + (size_t)(b * NBOX + n) * 5;
    float cls = t[0];
    float gx = t[1] * WW, gy = t[2] * HH, gw = t[3] * WW, gh = t[4] * HH;
    float valid = (t[1] > 0.0f) ? 1.0f : 0.0f;

    int best_a = 0; float best = -1.0f;
    for (int a = 0; a < AN; ++a) {
        float aw = d_anchors[a][0], ah = d_anchors[a][1];
        float inter = fminf(gw, aw) * fminf(gh, ah);
        float uni = gw * gh + aw * ah - inter;
        float r = inter / uni;
        if (r > best) { best = r; best_a = a; }   // first-max wins (argmax)
    }
    int gi = min(max((int)gx, 0), WW - 1);
    int gj = min(max((int)gy, 0), HH - 1);
    float aw = d_anchors[best_a][0], ah = d_anchors[best_a][1];
    float tx = gx - (float)gi, ty = gy - (float)gj;
    float tw = logf(gw / aw),  th = logf(gh / ah);

    size_t base = (size_t)(b * AN + best_a) * CHN * PLANE + (size_t)gj * WW + gi;
    float p0 = pred[base];
    float p1 = pred[base + PLANE];
    float p2 = pred[base + 2 * PLANE];
    float p3 = pred[base + 3 * PLANE];
    float bx = sigm(p0) + (float)gi;
    float by = sigm(p1) + (float)gj;
    float bw = expf(p2) * aw;
    float bh = expf(p3) * ah;
    float ig = iou_center(gx, gy, gw, gh, bx, by, bw, bh);

    float* r = gt + (size_t)(b * NBOX + n) * RECF;
    r[0] = valid; r[1] = (float)best_a; r[2] = (float)gi; r[3] = (float)gj;
    r[4] = gx; r[5] = gy; r[6] = gw; r[7] = gh;
    r[8] = tx; r[9] = ty; r[10] = tw; r[11] = th;
    r[12] = ig; r[13] = cls; r[14] = 0.0f; r[15] = 0.0f;
}

// ---- kernel 2: fused streaming loss over (b, a) planes
__global__ void k_main(const float* __restrict__ pred,
                       const float* __restrict__ gt,
                       float* __restrict__ acc) {
    int b = blockIdx.x / AN, a = blockIdx.x % AN;
    int tid = threadIdx.x;
    __shared__ float g[NBOX * RECF];
    const float* gsrc = gt + (size_t)b * NBOX * RECF;

    // Stage GT table into LDS via CDNA5 async-to-LDS when available.
#if defined(__gfx1250__) && __has_builtin(__builtin_amdgcn_global_load_async_to_lds_b32)
    for (int idx = tid; idx < NBOX * RECF; idx += (int)blockDim.x) {
        __builtin_amdgcn_global_load_async_to_lds_b32(
            (__attribute__((address_space(1))) int*)(gsrc + idx),
            (__attribute__((address_space(3))) int*)(&g[idx]),
            0, 0);
    }
#if __has_builtin(__builtin_amdgcn_s_wait_asynccnt)
    __builtin_amdgcn_s_wait_asynccnt(0);
#else
    asm volatile("s_wait_asynccnt 0x0" ::: "memory");
#endif
#else
    for (int idx = tid; idx < NBOX * RECF; idx += (int)blockDim.x)
        g[idx] = gsrc[idx];
#endif
    __syncthreads();

    float aw = d_anchors[a][0], ah = d_anchors[a][1];
    float sx = 0.f, sy = 0.f, sw = 0.f, sh = 0.f, sc = 0.f, scl = 0.f;
    size_t base_ba = (size_t)(b * AN + a) * CHN * PLANE;

    for (int cell = tid; cell < PLANE; cell += (int)blockDim.x) {
        int i = cell % WW, j = cell / WW;
        const float* pc = pred + base_ba + cell;
        float p0 = pc[0];
        float p1 = pc[PLANE];
        float p2 = pc[2 * PLANE];
        float p3 = pc[3 * PLANE];
        float p4 = pc[4 * PLANE];
        float s0 = sigm(p0), s1 = sigm(p1), conf = sigm(p4);
        float bx = s0 + (float)i, by = s1 + (float)j;
        float bw = expf(p2) * aw,  bh = expf(p3) * ah;

        float maxiou = 0.0f; int sn = -1;
        for (int n = 0; n < NBOX; ++n) {
            const float* r = &g[n * RECF];
            if (r[0] != 0.0f) {
                float v = iou_center(r[4], r[5], r[6], r[7], bx, by, bw, bh);
                maxiou = fmaxf(maxiou, v);
                if ((int)r[1] == a && (int)r[2] == i && (int)r[3] == j) sn = n;
            }
        }
        float cmask = (maxiou > 0.6f) ? 0.0f : 1.0f;   // NO_OBJECT_SCALE = 1
        float tb0 = 0.5f, tb1 = 0.5f, tb2 = 0.f, tb3 = 0.f, tconf = 0.f;
        if (sn >= 0) {
            const float* r = &g[sn * RECF];
            tb0 = r[8]; tb1 = r[9]; tb2 = r[10]; tb3 = r[11];
            tconf = r[12]; cmask = 5.0f;               // OBJECT_SCALE
            // cross-entropy over 80 class logits (online logsumexp)
            const float* cl = pc + 5 * PLANE;
            float m = cl[0]; float s = 1.0f;
            for (int c = 1; c < NCLS; ++c) {
                float l = cl[(size_t)c * PLANE];
                if (l > m) { s = s * expf(m - l) + 1.0f; m = l; }
                else       { s += expf(l - m); }
            }
            int tc = min(max((int)r[13], 0), NCLS - 1);
            float lt = cl[(size_t)tc * PLANE];
            scl += (m + logf(s)) - lt;
        }
        float dx = s0 - tb0, dy = s1 - tb1, dw2 = p2 - tb2, dh2 = p3 - tb3;
        float dc = conf - tconf;
        sx += dx * dx; sy += dy * dy; sw += dw2 * dw2; sh += dh2 * dh2;
        sc += dc * dc * cmask;
    }
    sx *= 0.5f; sy *= 0.5f; sw *= 0.5f; sh *= 0.5f; sc *= 0.5f;

    // per-wave reductions on the matrix pipe (EXEC all-ones here: no early exits)
    sx  = wave_sum(sx);  sy  = wave_sum(sy);
    sw  = wave_sum(sw);  sh  = wave_sum(sh);
    sc  = wave_sum(sc);  scl = wave_sum(scl);

    if ((tid & 31) == 0) {
        atomicAdd(&acc[0], sx);  atomicAdd(&acc[1], sy);
        atomicAdd(&acc[2], sw);  atomicAdd(&acc[3], sh);
        atomicAdd(&acc[4], sc);  atomicAdd(&acc[5], scl);
    }
}

// ---- kernel 3: finalize into d_out
__global__ void k_fin(const float* __restrict__ acc, float* __restrict__ out,
                      int full) {
    if (threadIdx.x == 0 && blockIdx.x == 0) {
        float lx = acc[0], ly = acc[1], lw = acc[2], lh = acc[3];
        float lc = acc[4], lcls = acc[5];
        float loss = lx + ly + lw + lh + lc + lcls;
        out[0] = loss;
        if (full) {
            out[1] = lx; out[2] = ly; out[3] = lw; out[4] = lh;
            out[5] = lc; out[6] = lcls;
        }
    }
}

extern "C" void kernel_launch(void* const* d_in, const int* in_sizes, int n_in,
                              void* d_out, int out_size, void* d_ws, size_t ws_size,
                              hipStream_t stream) {
    (void)in_sizes; (void)n_in; (void)ws_size;
    const float* pred   = (const float*)d_in[0];
    const float* target = (const float*)d_in[1];
    float* gt  = (float*)d_ws;                    // BB*NBOX*RECF floats (~100 KB)
    float* acc = gt + (size_t)BB * NBOX * RECF;   // 8 floats of accumulators

    k_zero<<<1, 32, 0, stream>>>(acc);
    k_pre<<<BB, 64, 0, stream>>>(pred, target, gt);
    k_main<<<BB * AN, 256, 0, stream>>>(pred, gt, acc);
    k_fin<<<1, 1, 0, stream>>>(acc, (float*)d_out, (out_size >= 7) ? 1 : 0);
}